// Embedding_34686155882936
// MI455X (gfx1250) — compile-verified
//
#include <hip/hip_runtime.h>
#include <stdint.h>

// ---------------------------------------------------------------------------
// Embedding gather: out[row, :] = table[ids[row], :]
//   rows   = BATCH*SEQ = 16384
//   D      = 768 fp32  = 3072 bytes/row = 192 float4 = 6 float4 per lane (wave32)
// Memory-bound: ~100 MB total traffic -> ~4.3us floor @ 23.3 TB/s.
// CDNA5 path: async DMA global -> LDS -> global (ASYNCcnt), VGPR-free staging.
// Fallback:   B128 loads (table stays L2-resident, 154MB < 192MB L2) +
//             non-temporal B128 stores (don't evict the table from L2).
// ---------------------------------------------------------------------------

#define AS1 __attribute__((address_space(1)))
#define AS3 __attribute__((address_space(3)))

#if __has_builtin(__builtin_amdgcn_global_load_async_to_lds_b128) && \
    __has_builtin(__builtin_amdgcn_global_store_async_from_lds_b128)
#define EMB_USE_ASYNC 1
#else
#define EMB_USE_ASYNC 0
#endif

static __device__ __forceinline__ void wait_async_0() {
#if __has_builtin(__builtin_amdgcn_s_wait_asynccnt)
  __builtin_amdgcn_s_wait_asynccnt(0);
#else
  asm volatile("s_wait_asynccnt 0" ::: "memory");
#endif
}

typedef float f4  __attribute__((ext_vector_type(4)));
typedef int   v4i __attribute__((ext_vector_type(4)));   // 16B unit for async B128

constexpr int kD        = 768;          // floats per row
constexpr int kRowBytes = kD * 4;       // 3072 bytes
constexpr int kWaves    = 8;            // waves per block
constexpr int kBlock    = kWaves * 32;  // 256 threads

__global__ __launch_bounds__(kBlock)
void embedding_gather_kernel(const int* __restrict__ ids,
                             const float* __restrict__ table,
                             float* __restrict__ out,
                             int nrows)
{
  const int lane = threadIdx.x & 31;
  const int wid  = threadIdx.x >> 5;
  const int row  = blockIdx.x * kWaves + wid;   // one wave32 per output row
  if (row >= nrows) return;                     // wave-uniform, EXEC stays all-1s

  // Token id is wave-uniform -> force into SGPR so addressing is scalar.
  int tok = __builtin_amdgcn_readfirstlane(ids[row]);

  const float* src = table + (size_t)tok * kD;  // gather source row
  float*       dst = out   + (size_t)row * kD;  // streaming destination row

#if EMB_USE_ASYNC
  // --- CDNA5 async DMA path: global -> LDS -> global, no VGPR data movement.
  __shared__ unsigned char smem[kWaves * kRowBytes];   // 24 KB / block

  // Generic->LDS pointer: low 32 bits of a generic LDS address are the LDS
  // offset (aperture rule), so route the cast through an integer.
  AS3 v4i* l = (AS3 v4i*)(unsigned int)(uintptr_t)&smem[wid * kRowBytes + lane * 16];
  AS1 v4i* g = (AS1 v4i*)(uintptr_t)(src + lane * 4);
  AS1 v4i* o = (AS1 v4i*)(uintptr_t)(dst + lane * 4);

  // 6 x (32 lanes x 16B) = 3072B per wave. Stride 512B = 32 v4i elements.
  __builtin_amdgcn_global_load_async_to_lds_b128(g,       l,       0, 0);
  __builtin_amdgcn_global_load_async_to_lds_b128(g + 32,  l + 32,  0, 0);
  __builtin_amdgcn_global_load_async_to_lds_b128(g + 64,  l + 64,  0, 0);
  __builtin_amdgcn_global_load_async_to_lds_b128(g + 96,  l + 96,  0, 0);
  __builtin_amdgcn_global_load_async_to_lds_b128(g + 128, l + 128, 0, 0);
  __builtin_amdgcn_global_load_async_to_lds_b128(g + 160, l + 160, 0, 0);

  // Async loads/stores share ASYNCcnt but LDS-side ordering between a load's
  // LDS write and a store's LDS read is not guaranteed -> drain loads first.
  wait_async_0();

  __builtin_amdgcn_global_store_async_from_lds_b128(o,       l,       0, 0);
  __builtin_amdgcn_global_store_async_from_lds_b128(o + 32,  l + 32,  0, 0);
  __builtin_amdgcn_global_store_async_from_lds_b128(o + 64,  l + 64,  0, 0);
  __builtin_amdgcn_global_store_async_from_lds_b128(o + 96,  l + 96,  0, 0);
  __builtin_amdgcn_global_store_async_from_lds_b128(o + 128, l + 128, 0, 0);
  __builtin_amdgcn_global_store_async_from_lds_b128(o + 160, l + 160, 0, 0);
  // S_ENDPGM performs an implicit wait-idle, draining the async stores.
#else
  // --- Fallback: B128 register gather.
  // Table loads: default (RT) temporal hint -> table rows stay resident in the
  // 192 MB L2 (table is 154 MB). Output: non-temporal stores so the 50 MB
  // stream does not evict the table.
  const f4* s4 = (const f4*)src + lane;
  f4*       d4 = (f4*)dst + lane;
#pragma unroll
  for (int k = 0; k < 6; ++k) {
    f4 v = s4[k * 32];
    __builtin_nontemporal_store(v, &d4[k * 32]);
  }
#endif
}

extern "C" void kernel_launch(void* const* d_in, const int* in_sizes, int n_in,
                              void* d_out, int out_size, void* d_ws, size_t ws_size,
                              hipStream_t stream) {
  const int*   ids   = (const int*)d_in[0];     // token_ids, flat (B*S)
  const float* table = (const float*)d_in[1];   // [VOCAB, 768] fp32
  float*       out   = (float*)d_out;           // [B*S, 768] fp32

  const int nrows = in_sizes[0];                // 16384
  const int grid  = (nrows + kWaves - 1) / kWaves;

  embedding_gather_kernel<<<grid, kBlock, 0, stream>>>(ids, table, out, nrows);
}